// GsLmkEncoder_64733747085511
// MI455X (gfx1250) — compile-verified
//
#include <hip/hip_runtime.h>
#include <hip/hip_bf16.h>

typedef __attribute__((ext_vector_type(2))) float v2f;
typedef __attribute__((ext_vector_type(8))) float v8f;

#define LNUM 68
#define KDIM 12
#define COLS 80          // 68 landmarks padded to 5 tiles of 16
#define OUTD 340         // 5 * 68

// ---------------------------------------------------------------------------
// Prep kernel: build per-landmark coefficient matrices Bwq[12][80], Bdz[12][80]
// wq(n,l) = f(x_n) . Bwq[:,l],  dz(n,l) = f(x_n) . Bdz[:,l]
// with f(x) = [x^2, y^2, z^2, xy, xz, yz, x, y, z, 1, 0, 0]
// ---------------------------------------------------------------------------
__global__ void gs_prep(const float* __restrict__ lmk, const float* __restrict__ r,
                        const float* __restrict__ scal, const float* __restrict__ rot,
                        float* __restrict__ Bwq, float* __restrict__ Bdz) {
  int tid = threadIdx.x;
  for (int i = tid; i < KDIM * COLS; i += blockDim.x) { Bwq[i] = 0.0f; Bdz[i] = 0.0f; }
  __syncthreads();
  if (tid >= LNUM) return;
  const int l = tid;

  // normalized quaternion -> rotation matrix
  float qw = rot[l*4+0], qx = rot[l*4+1], qy = rot[l*4+2], qz = rot[l*4+3];
  float nrm = fmaxf(sqrtf(qw*qw + qx*qx + qy*qy + qz*qz), 1e-12f);
  qw /= nrm; qx /= nrm; qy /= nrm; qz /= nrm;
  float R00 = 1.f - 2.f*(qy*qy + qz*qz), R01 = 2.f*(qx*qy - qw*qz), R02 = 2.f*(qx*qz + qw*qy);
  float R10 = 2.f*(qx*qy + qw*qz), R11 = 1.f - 2.f*(qx*qx + qz*qz), R12 = 2.f*(qy*qz - qw*qx);
  float R20 = 2.f*(qx*qz - qw*qy), R21 = 2.f*(qy*qz + qw*qx), R22 = 1.f - 2.f*(qx*qx + qy*qy);

  // M = R with column j divided by scaling[j]; cov = M * M^T (symmetric)
  float s0 = scal[l*3+0], s1 = scal[l*3+1], s2 = scal[l*3+2];
  float M00 = R00/s0, M01 = R01/s1, M02 = R02/s2;
  float M10 = R10/s0, M11 = R11/s1, M12 = R12/s2;
  float M20 = R20/s0, M21 = R21/s1, M22 = R22/s2;
  float c00 = M00*M00 + M01*M01 + M02*M02;
  float c01 = M00*M10 + M01*M11 + M02*M12;
  float c02 = M00*M20 + M01*M21 + M02*M22;
  float c11 = M10*M10 + M11*M11 + M12*M12;
  float c12 = M10*M20 + M11*M21 + M12*M22;
  float c22 = M20*M20 + M21*M21 + M22*M22;

  float mx = lmk[l*3+0], my = lmk[l*3+1], mz = lmk[l*3+2];
  // fold (x-mu)^T cov (x-mu) into quadratic/linear/constant terms of x
  float gx = c00*mx + c01*my + c02*mz;
  float gy = c01*mx + c11*my + c12*mz;
  float gz = c02*mx + c12*my + c22*mz;

  Bwq[0*COLS+l] = c00;       Bwq[1*COLS+l] = c11;       Bwq[2*COLS+l] = c22;
  Bwq[3*COLS+l] = 2.f*c01;   Bwq[4*COLS+l] = 2.f*c02;   Bwq[5*COLS+l] = 2.f*c12;
  Bwq[6*COLS+l] = -2.f*gx;   Bwq[7*COLS+l] = -2.f*gy;   Bwq[8*COLS+l] = -2.f*gz;
  Bwq[9*COLS+l] = mx*gx + my*gy + mz*gz;

  // dz = r_z . (x - mu);  r is row-major 3x3, third column
  float rz0 = r[2], rz1 = r[5], rz2 = r[8];
  Bdz[6*COLS+l] = rz0; Bdz[7*COLS+l] = rz1; Bdz[8*COLS+l] = rz2;
  Bdz[9*COLS+l] = -(rz0*mx + rz1*my + rz2*mz);
}

// exp(-0.5*wq) as a single v_exp_f32; sin/cos of dz; double-angle for 2*dz.
__device__ __forceinline__ void gs_emit(float wq, float dz, float* o) {
  const float w  = __builtin_amdgcn_exp2f(wq * -0.72134752044448170f); // -0.5*log2(e)
  const float s1 = __sinf(dz);
  const float c1 = __cosf(dz);
  const float s2 = 2.0f * s1 * c1;                        // sin(2dz)
  const float c2 = 1.0f - 2.0f * s1 * s1;                 // cos(2dz)
  o[0]        = dz * w;
  o[LNUM]     = s1 * w;
  o[2*LNUM]   = c1 * w;
  o[3*LNUM]   = s2 * w;
  o[4*LNUM]   = c2 * w;
}

// ---------------------------------------------------------------------------
// Main kernel: each wave32 computes a 16-point x 80-landmark stripe via
// V_WMMA_F32_16X16X4_F32 (3 K-steps of K=4 cover KDIM=12), for both wq and dz
// with identical column mapping, then exp/sin/cos + stores with constant
// immediate offsets (one divergent guard per tile, not per element).
// ---------------------------------------------------------------------------
__global__ __launch_bounds__(256) void gs_main(const float* __restrict__ x,
                                               const float* __restrict__ Bwq,
                                               const float* __restrict__ Bdz,
                                               float* __restrict__ out, int N) {
  const int lane = threadIdx.x & 31;
  const int wave = threadIdx.x >> 5;
  const int gw   = blockIdx.x * 8 + wave;          // global wave index
  if (gw * 16 >= N) return;                         // wave-uniform: EXEC stays full
  const int base_n = gw * 16;
  const bool fullwave = (base_n + 16 <= N);         // wave-uniform (always true here)
  const int hv   = lane >> 4;                       // wave half: 0 or 1
  const int mrow = lane & 15;

  // Per-lane point features (row M = mrow of the A tile)
  int n0 = base_n + mrow; if (n0 >= N) n0 = N - 1;  // clamp (N is 16-aligned anyway)
  const float px = x[n0*3+0], py = x[n0*3+1], pz = x[n0*3+2];
  const float f0 = px*px, f1 = py*py, f2 = pz*pz;
  const float f3 = px*py, f4 = px*pz, f5 = py*pz;
  const float f6 = px,    f7 = py,    f8 = pz, f9 = 1.0f, f10 = 0.0f, f11 = 0.0f;

  // A fragments: 16x4 f32 layout -> VGPR0 holds K = 4kk + 2*hv, VGPR1 the next K
  v2f a0, a1, a2;
  a0.x = hv ? f2  : f0;  a0.y = hv ? f3  : f1;
  a1.x = hv ? f6  : f4;  a1.y = hv ? f7  : f5;
  a2.x = hv ? f10 : f8;  a2.y = hv ? f11 : f9;

  const int kb = 2 * hv;   // B-fragment K base within each K-step

  for (int t = 0; t < 5; ++t) {
    const int col = t * 16 + mrow;                  // landmark (column) index

    v2f bq0 = { Bwq[(0+kb)*COLS+col], Bwq[(1+kb)*COLS+col] };
    v2f bq1 = { Bwq[(4+kb)*COLS+col], Bwq[(5+kb)*COLS+col] };
    v2f bq2 = { Bwq[(8+kb)*COLS+col], Bwq[(9+kb)*COLS+col] };
    v2f bd0 = { Bdz[(0+kb)*COLS+col], Bdz[(1+kb)*COLS+col] };
    v2f bd1 = { Bdz[(4+kb)*COLS+col], Bdz[(5+kb)*COLS+col] };
    v2f bd2 = { Bdz[(8+kb)*COLS+col], Bdz[(9+kb)*COLS+col] };

    v8f cq = {};
    v8f cd = {};
    cq = __builtin_amdgcn_wmma_f32_16x16x4_f32(false, a0, false, bq0, (short)0, cq, false, false);
    cq = __builtin_amdgcn_wmma_f32_16x16x4_f32(false, a1, false, bq1, (short)0, cq, false, false);
    cq = __builtin_amdgcn_wmma_f32_16x16x4_f32(false, a2, false, bq2, (short)0, cq, false, false);
    cd = __builtin_amdgcn_wmma_f32_16x16x4_f32(false, a0, false, bd0, (short)0, cd, false, false);
    cd = __builtin_amdgcn_wmma_f32_16x16x4_f32(false, a1, false, bd1, (short)0, cd, false, false);
    cd = __builtin_amdgcn_wmma_f32_16x16x4_f32(false, a2, false, bd2, (short)0, cd, false, false);

    // D tile layout: VGPR r -> row M = r + hv*8, column N = col.
    if (col < LNUM) {                               // one divergent guard per tile
      if (fullwave) {
        // constant-offset stores: o + r*340 + b*68
        float* o = out + (size_t)(base_n + hv * 8) * OUTD + col;
#pragma unroll
        for (int r = 0; r < 8; ++r)
          gs_emit(cq[r], cd[r], o + r * OUTD);
      } else {
        // partial tail wave: clamped rows recompute row N-1's exact values,
        // so the rewrite is deterministic and in-bounds.
#pragma unroll
        for (int r = 0; r < 8; ++r) {
          int n = base_n + hv * 8 + r; if (n >= N) n = N - 1;
          gs_emit(cq[r], cd[r], out + (size_t)n * OUTD + col);
        }
      }
    }
  }
}

extern "C" void kernel_launch(void* const* d_in, const int* in_sizes, int n_in,
                              void* d_out, int out_size, void* d_ws, size_t ws_size,
                              hipStream_t stream) {
  const float* x    = (const float*)d_in[0];   // (N,3)
  const float* lmk  = (const float*)d_in[1];   // (L,3)
  const float* r    = (const float*)d_in[2];   // (3,3)
  const float* scal = (const float*)d_in[3];   // (L,3)
  const float* rot  = (const float*)d_in[4];   // (L,4)
  float* out = (float*)d_out;

  float* ws  = (float*)d_ws;
  float* Bwq = ws;
  float* Bdz = ws + KDIM * COLS;

  gs_prep<<<1, 128, 0, stream>>>(lmk, r, scal, rot, Bwq, Bdz);

  const int N = in_sizes[0] / 3;
  const int waves  = (N + 15) / 16;
  const int blocks = (waves + 7) / 8;              // 8 wave32 per 256-thread block
  gs_main<<<blocks, 256, 0, stream>>>(x, Bwq, Bdz, out, N);
}